// DisLoss_17325898072321
// MI455X (gfx1250) — compile-verified
//
#include <hip/hip_runtime.h>
#include <math.h>

#define N_CLS 8192
#define FEAT  512
#define BATCH 1024
#define INV_T 10.0f            // 1 / TEMPERATURE
#define EMA_M 0.95f

typedef float v2f __attribute__((ext_vector_type(2)));
typedef float v8f __attribute__((ext_vector_type(8)));

// ---------------------------------------------------------------------------
// Kernel 1: per-label sequential EMA prototype update (+ copy into workspace)
// ---------------------------------------------------------------------------
__global__ __launch_bounds__(512)
void ema_kernel(const float* __restrict__ features,
                const int*   __restrict__ labels,
                const float* __restrict__ protos_in,
                float*       __restrict__ P)
{
    const int L = blockIdx.x;
    const int t = threadIdx.x;
    __shared__ float red[16];

    float p = protos_in[(size_t)L * FEAT + t];

    for (int i = 0; i < BATCH; ++i) {
        const int li = labels[i];
        if (li == L) {                 // uniform branch within block
            float f = features[(size_t)i * FEAT + t];
            float v = p * EMA_M + f * (1.0f - EMA_M);
            float s = v * v;
            s += __shfl_xor(s, 1,  32);
            s += __shfl_xor(s, 2,  32);
            s += __shfl_xor(s, 4,  32);
            s += __shfl_xor(s, 8,  32);
            s += __shfl_xor(s, 16, 32);
            if ((t & 31) == 0) red[t >> 5] = s;
            __syncthreads();
            float tot = 0.0f;
            #pragma unroll
            for (int w = 0; w < 16; ++w) tot += red[w];
            const float inv = 1.0f / fmaxf(sqrtf(tot), 1e-12f);
            p = v * inv;
            __syncthreads();
        }
    }
    P[(size_t)L * FEAT + t] = p;
}

// ---------------------------------------------------------------------------
// Kernel 2: fused P @ P^T -> exp(logit/T) -> off-diagonal partial row sums.
// Double-buffered LDS with GLOBAL_LOAD_ASYNC_TO_LDS_B128 prefetch (ASYNCcnt):
// issue next chunk's DMA right after the barrier, overlap with 64 WMMAs.
// ---------------------------------------------------------------------------
#define BM 64           // rows owned per block
#define BN 128          // column tile per iteration
#define KC 64           // K chunk staged in LDS
#define SA (KC + 4)     // 68: LDS row stride (68 mod 64 = 4 -> conflict-free)
#define NSPLIT 4
#define NCHUNK (N_CLS / NSPLIT)      // 2048 columns per split
#define NKCH   (FEAT / KC)           // 8 K-chunks
#define NITER  ((NCHUNK / BN) * NKCH)
#define BUFSZ  ((BM + BN) * SA)      // floats per LDS buffer

#define WMMA_F32(A, B, C) \
    __builtin_amdgcn_wmma_f32_16x16x4_f32(false, (A), false, (B), (short)0, (C), false, false)

__device__ __forceinline__ void async_ld_b128(float* lds, const float* gsrc)
{
    // per-lane: copies 16 bytes from the global address in vaddr to the LDS address in vdst, tracked with ASYNCcnt
    const unsigned lds_off = (unsigned)(size_t)lds;   // low 32 bits = LDS offset
    asm volatile("global_load_async_to_lds_b128 %0, %1, off"
                 :: "v"(lds_off), "v"(gsrc) : "memory");
}

__device__ __forceinline__ void wait_async0()
{
#if __has_builtin(__builtin_amdgcn_s_wait_asynccnt)
    __builtin_amdgcn_s_wait_asynccnt(0);
#else
    asm volatile("s_wait_asynccnt 0x0" ::: "memory");
#endif
}

__global__ __launch_bounds__(256)
void disp_kernel(const float* __restrict__ P,
                 float*       __restrict__ partial)   // [NSPLIT][N_CLS]
{
    extern __shared__ float smem[];                   // 2 * BUFSZ floats

    const int tid    = threadIdx.x;
    const int lane   = tid & 31;
    const int waveId = tid >> 5;
    const int wm     = waveId >> 2;    // 0..1
    const int wn     = waveId & 3;     // 0..3
    const int lr     = lane & 15;
    const int hi     = lane >> 4;
    const int koff   = hi * 2;

    const int mBase      = blockIdx.x * BM;
    const int nSplitBase = blockIdx.y * NCHUNK;

    float rowAcc[16];
    #pragma unroll
    for (int v = 0; v < 16; ++v) rowAcc[v] = 0.0f;

    v8f acc00 = {0.f,0.f,0.f,0.f,0.f,0.f,0.f,0.f};
    v8f acc01 = acc00, acc10 = acc00, acc11 = acc00;

    // ---- async-issue one (nt, kc) chunk into LDS buffer `buf` ----
    auto issue_chunk = [&](int it, int buf) {
        const int nt    = it >> 3;                 // / NKCH
        const int kc    = (it & (NKCH - 1)) * KC;
        const int nBase = nSplitBase + nt * BN;
        float* As_ = smem + buf * BUFSZ;
        float* Bs_ = As_ + BM * SA;
        // A chunk: 64 x 64 -> 1024 float4 (4 per thread)
        for (int idx = tid; idx < BM * (KC / 4); idx += 256) {
            const int row = idx >> 4;
            const int col = (idx & 15) * 4;
            async_ld_b128(&As_[row * SA + col],
                          &P[(size_t)(mBase + row) * FEAT + kc + col]);
        }
        // B chunk: 128 x 64 -> 2048 float4 (8 per thread)
        for (int idx = tid; idx < BN * (KC / 4); idx += 256) {
            const int row = idx >> 4;
            const int col = (idx & 15) * 4;
            async_ld_b128(&Bs_[row * SA + col],
                          &P[(size_t)(nBase + row) * FEAT + kc + col]);
        }
    };

    issue_chunk(0, 0);                 // prologue prefetch

    for (int it = 0; it < NITER; ++it) {
        const int buf = it & 1;
        // my batch for `buf` is the only outstanding one -> wait, then barrier:
        // after it, everyone's data is in LDS and buf^1 is free to overwrite.
        wait_async0();
        __syncthreads();
        if (it + 1 < NITER) issue_chunk(it + 1, buf ^ 1);   // overlap with compute

        const float* As_ = smem + buf * BUFSZ;
        const float* Bs_ = As_ + BM * SA;
        const float* a0  = &As_[(wm * 32 + lr) * SA] + koff;
        const float* a1  = a0 + 16 * SA;
        const float* b0  = &Bs_[(wn * 32 + lr) * SA] + koff;
        const float* b1  = b0 + 16 * SA;

        #pragma unroll
        for (int kk = 0; kk < KC; kk += 4) {
            const v2f A0 = *(const v2f*)(a0 + kk);
            const v2f A1 = *(const v2f*)(a1 + kk);
            const v2f B0 = *(const v2f*)(b0 + kk);
            const v2f B1 = *(const v2f*)(b1 + kk);
            acc00 = WMMA_F32(A0, B0, acc00);   // 4 independent chains
            acc10 = WMMA_F32(A1, B0, acc10);
            acc01 = WMMA_F32(A0, B1, acc01);
            acc11 = WMMA_F32(A1, B1, acc11);
        }

        // ---- epilogue at the end of each column tile (last K chunk) ----
        if ((it & (NKCH - 1)) == (NKCH - 1)) {
            const int nBase = nSplitBase + (it >> 3) * BN;
            const int gn0 = nBase + wn * 32 + lr;
            const int gn1 = gn0 + 16;
            #pragma unroll
            for (int v = 0; v < 8; ++v) {
                const int gm0 = mBase + wm * 32 + v + 8 * hi;
                const int gm1 = gm0 + 16;
                float e00 = __expf(acc00[v] * INV_T); if (gm0 == gn0) e00 = 0.0f;
                float e01 = __expf(acc01[v] * INV_T); if (gm0 == gn1) e01 = 0.0f;
                float e10 = __expf(acc10[v] * INV_T); if (gm1 == gn0) e10 = 0.0f;
                float e11 = __expf(acc11[v] * INV_T); if (gm1 == gn1) e11 = 0.0f;
                rowAcc[v]     += e00 + e01;
                rowAcc[8 + v] += e10 + e11;
                acc00[v] = 0.0f; acc01[v] = 0.0f;
                acc10[v] = 0.0f; acc11[v] = 0.0f;
            }
        }
    }

    // ---- reduce rowAcc over the 16 lanes of each half-wave ----
    #pragma unroll
    for (int r = 0; r < 16; ++r) {
        float s = rowAcc[r];
        s += __shfl_xor(s, 1, 32);
        s += __shfl_xor(s, 2, 32);
        s += __shfl_xor(s, 4, 32);
        s += __shfl_xor(s, 8, 32);
        rowAcc[r] = s;                 // valid where lr == 0
    }

    __syncthreads();                   // LDS free for reuse
    float* pbuf = smem;                // [64][4]
    if (lr == 0) {
        #pragma unroll
        for (int r = 0; r < 16; ++r) {
            const int rl = wm * 32 + ((r >> 3) * 16) + (r & 7) + 8 * hi; // 0..63
            pbuf[rl * 4 + wn] = rowAcc[r];
        }
    }
    __syncthreads();
    if (tid < BM) {
        const float s = pbuf[tid * 4 + 0] + pbuf[tid * 4 + 1] +
                        pbuf[tid * 4 + 2] + pbuf[tid * 4 + 3];
        partial[(size_t)blockIdx.y * N_CLS + mBase + tid] = s;  // plain store
    }
}

// ---------------------------------------------------------------------------
// Kernel 3: rowsum_i = sum_s partial[s][i];  loss = mean_i log(rowsum/(n-1))
// ---------------------------------------------------------------------------
__global__ __launch_bounds__(256)
void loss_kernel(const float* __restrict__ partial, float* __restrict__ out)
{
    __shared__ float red[8];
    const int tid = threadIdx.x;
    float s = 0.0f;
    for (int i = tid; i < N_CLS; i += 256) {
        float rs = 0.0f;
        #pragma unroll
        for (int sp = 0; sp < NSPLIT; ++sp)
            rs += partial[(size_t)sp * N_CLS + i];
        s += __logf(rs * (1.0f / (float)(N_CLS - 1)));
    }
    s += __shfl_xor(s, 1,  32);
    s += __shfl_xor(s, 2,  32);
    s += __shfl_xor(s, 4,  32);
    s += __shfl_xor(s, 8,  32);
    s += __shfl_xor(s, 16, 32);
    if ((tid & 31) == 0) red[tid >> 5] = s;
    __syncthreads();
    if (tid == 0) {
        float t = 0.0f;
        #pragma unroll
        for (int w = 0; w < 8; ++w) t += red[w];
        out[0] = t / (float)N_CLS;
    }
}

// ---------------------------------------------------------------------------
extern "C" void kernel_launch(void* const* d_in, const int* in_sizes, int n_in,
                              void* d_out, int out_size, void* d_ws, size_t ws_size,
                              hipStream_t stream)
{
    const float* features = (const float*)d_in[0];   // [1024, 512] f32
    const int*   labels   = (const int*)  d_in[1];   // [1024] i32
    const float* protos   = (const float*)d_in[2];   // [8192, 512] f32

    float* P       = (float*)d_ws;                   // updated prototypes copy
    float* partial = P + (size_t)N_CLS * FEAT;       // [NSPLIT][8192]

    ema_kernel<<<N_CLS, 512, 0, stream>>>(features, labels, protos, P);

    const size_t shmem = (size_t)(2 * BUFSZ) * sizeof(float);   // ~104 KB
    dim3 grid(N_CLS / BM, NSPLIT);
    disp_kernel<<<grid, 256, shmem, stream>>>(P, partial);

    loss_kernel<<<1, 256, 0, stream>>>(partial, (float*)d_out);
}